// GCNConv_4861902979730
// MI455X (gfx1250) — compile-verified
//
#include <hip/hip_runtime.h>

typedef __attribute__((ext_vector_type(2))) float v2f;
typedef __attribute__((ext_vector_type(4))) float v4f;
typedef __attribute__((ext_vector_type(8))) float v8f;

#define N_NODES 50000
#define DEG 16
#define D 64
#define ROWS_PER_BLOCK 32
#define LDS_STRIDE 68   // 68 mod 64 = 4 -> conflict-free fragment reads

// Fused GCNConv: H = A @ X (gather-sum, exact fp32), then tileout = H @ W via
// V_WMMA_F32_16X16X4_F32 (fp32 matrix pipe, bit-compatible accumulation).
__global__ __launch_bounds__(256)
void gcn_fused_wmma_kernel(const float* __restrict__ X,
                           const float* __restrict__ W,
                           const int*   __restrict__ row_pointers,
                           const int*   __restrict__ column_index,
                           float*       __restrict__ out)
{
    __shared__ float Hs[ROWS_PER_BLOCK * LDS_STRIDE]; // aggregated rows
    __shared__ float Wt[D * LDS_STRIDE];              // W transposed: Wt[n][k]

    const int t        = threadIdx.x;
    const int row_base = blockIdx.x * ROWS_PER_BLOCK;
    const bool full_tile = (row_base + ROWS_PER_BLOCK) <= N_NODES; // uniform

    // --- Stage W into LDS, transposed (coalesced global read) ---
    for (int idx = t; idx < D * D; idx += 256) {
        const int i = idx >> 6;   // k (row of W)
        const int j = idx & 63;   // n (col of W)
        Wt[j * LDS_STRIDE + i] = W[idx];
    }

    // --- Phase A: neighborhood aggregation H[r][:] = sum_j X[col[j]][:] ---
    {
        const int r  = t >> 3;          // local row 0..31
        const int cg = t & 7;           // 8 chunks of 8 floats
        const int gr = row_base + r;
        v4f acc0 = {0.f, 0.f, 0.f, 0.f};
        v4f acc1 = {0.f, 0.f, 0.f, 0.f};
        if (full_tile || gr < N_NODES) {
            const long eb = (long)row_pointers[gr]; // == gr*DEG (uniform degree)
            #pragma unroll
            for (int j = 0; j < DEG; ++j) {
                const int  c  = column_index[eb + j];
                const float* xr = X + (long)c * D + cg * 8;
                acc0 += *(const v4f*)(xr);
                acc1 += *(const v4f*)(xr + 4);
            }
        }
        *(v4f*)&Hs[r * LDS_STRIDE + cg * 8]     = acc0;
        *(v4f*)&Hs[r * LDS_STRIDE + cg * 8 + 4] = acc1;
    }
    __syncthreads();

    // --- Phase B: 8 waves x (2 row-windows x 4 col-tiles) of 16x16 output ---
    const int wave = t >> 5;        // 0..7
    const int lane = t & 31;
    const int rw   = wave >> 2;     // row window within block: 0/1
    const int ct   = wave & 3;      // 16-col tile: 0..3
    const int m    = lane & 15;     // M (A) / N (B,C,D) index
    const int half = lane >> 4;     // K-half selector

    v8f c = {};
    const float* ha = &Hs[(rw * 16 + m) * LDS_STRIDE];  // A row (M = m)
    const float* wb = &Wt[(ct * 16 + m) * LDS_STRIDE];  // B col (N = m), transposed
    #pragma unroll
    for (int kk = 0; kk < 16; ++kk) {
        // fp32 A 16x4: lane(m,h), VGPR v -> A[m][4*kk + 2h + v]  (contiguous pair)
        v2f a = *(const v2f*)(ha + kk * 4 + 2 * half);
        // fp32 B 4x16: lane(n,h), VGPR v -> B[4*kk + 2h + v][n] = Wt[n][...]
        v2f b = *(const v2f*)(wb + kk * 4 + 2 * half);
        c = __builtin_amdgcn_wmma_f32_16x16x4_f32(
                /*neg_a=*/false, a, /*neg_b=*/false, b,
                /*c_mod=*/(short)0, c, /*reuse_a=*/false, /*reuse_b=*/false);
    }

    // --- Store D tile: VGPR r -> M = r (lanes 0-15) / r+8 (lanes 16-31) ---
    // Block-uniform fast path: only the last block can have rows >= N_NODES,
    // so full tiles store with one base address + immediate offsets.
    const int gr0   = row_base + rw * 16;
    const int mrow0 = gr0 + half * 8;                     // first M row this lane writes
    float* op = out + (long)mrow0 * D + ct * 16 + m;
    if (full_tile) {
        #pragma unroll
        for (int r = 0; r < 8; ++r)
            op[r * D] = c[r];                             // imm offset r*256B
    } else {
        #pragma unroll
        for (int r = 0; r < 8; ++r)
            if (mrow0 + r < N_NODES)
                op[r * D] = c[r];
    }
}

extern "C" void kernel_launch(void* const* d_in, const int* in_sizes, int n_in,
                              void* d_out, int out_size, void* d_ws, size_t ws_size,
                              hipStream_t stream) {
    (void)in_sizes; (void)n_in; (void)out_size; (void)d_ws; (void)ws_size;
    const float* X            = (const float*)d_in[0];
    const float* W            = (const float*)d_in[1];
    const int*   row_pointers = (const int*)d_in[2];
    const int*   column_index = (const int*)d_in[3];
    float*       out          = (float*)d_out;

    const int grid = (N_NODES + ROWS_PER_BLOCK - 1) / ROWS_PER_BLOCK; // 1563
    gcn_fused_wmma_kernel<<<grid, 256, 0, stream>>>(X, W, row_pointers,
                                                    column_index, out);
}